// GCN_39281770889756
// MI455X (gfx1250) — compile-verified
//
#include <hip/hip_runtime.h>

typedef __attribute__((ext_vector_type(2))) float v2f;
typedef __attribute__((ext_vector_type(8))) float v8f;

#define NNODES 100000
#define DIM 64
#define NTILES_M (NNODES / 16)   // 6250, exact

// ---------------------------------------------------------------------------
// xw = X @ W  using V_WMMA_F32_16X16X4_F32.
// One wave computes a 16x64 output slab (4 column tiles), looping K in 4s.
// A 16x4 f32 frag: lanes 0-15 hold {K=0,K=1} for M=lane, lanes 16-31 {K=2,K=3}.
// B 4x16 f32 frag: lanes 0-15 hold {K=0,K=1} for N=lane, lanes 16-31 {K=2,K=3}.
// C/D 16x16 f32:   VGPR r -> M = r + 8*(lane>>4), N = lane&15.
// ---------------------------------------------------------------------------
__global__ __launch_bounds__(256) void gemm_wmma(const float* __restrict__ X,
                                                 const float* __restrict__ W,
                                                 float* __restrict__ Y) {
  const int wave  = blockIdx.x * (blockDim.x >> 5) + (threadIdx.x >> 5);
  const int lane  = threadIdx.x & 31;
  if (wave >= NTILES_M) return;           // whole wave exits together: EXEC stays all-1s

  const int mrow  = wave * 16 + (lane & 15);
  const int khalf = (lane >> 4) * 2;      // 0 for lanes 0-15, 2 for lanes 16-31
  const int nlo   = lane & 15;

  v8f acc0 = {}, acc1 = {}, acc2 = {}, acc3 = {};

  const float* xrow = X + (size_t)mrow * DIM;
  for (int k0 = 0; k0 < DIM; k0 += 4) {
    v2f a;
    a.x = xrow[k0 + khalf + 0];
    a.y = xrow[k0 + khalf + 1];
    const float* w0 = W + (size_t)(k0 + khalf) * DIM + nlo;
    v2f b0; b0.x = w0[0];        b0.y = w0[DIM];
    v2f b1; b1.x = w0[16];       b1.y = w0[DIM + 16];
    v2f b2; b2.x = w0[32];       b2.y = w0[DIM + 32];
    v2f b3; b3.x = w0[48];       b3.y = w0[DIM + 48];
    acc0 = __builtin_amdgcn_wmma_f32_16x16x4_f32(false, a, false, b0, (short)0, acc0, false, false);
    acc1 = __builtin_amdgcn_wmma_f32_16x16x4_f32(false, a, false, b1, (short)0, acc1, false, false);
    acc2 = __builtin_amdgcn_wmma_f32_16x16x4_f32(false, a, false, b2, (short)0, acc2, false, false);
    acc3 = __builtin_amdgcn_wmma_f32_16x16x4_f32(false, a, false, b3, (short)0, acc3, false, false);
  }

  const int rbase = wave * 16 + (lane >> 4) * 8;
  float* yb = Y + (size_t)rbase * DIM + nlo;
#pragma unroll
  for (int r = 0; r < 8; ++r) {
    yb[(size_t)r * DIM +  0] = acc0[r];
    yb[(size_t)r * DIM + 16] = acc1[r];
    yb[(size_t)r * DIM + 32] = acc2[r];
    yb[(size_t)r * DIM + 48] = acc3[r];
  }
}

// out[n][d] = RESIDUAL * ori[n][d] + bias[d]    (RESIDUAL == 1.0)
__global__ __launch_bounds__(256) void init_out(const float* __restrict__ ori,
                                                const float* __restrict__ bias,
                                                float* __restrict__ out) {
  const int i = blockIdx.x * blockDim.x + threadIdx.x;  // one float4 per thread
  if (i >= NNODES * (DIM / 4)) return;
  const float4 o = ((const float4*)ori)[i];
  const float4 b = ((const float4*)bias)[i & (DIM / 4 - 1)];
  float4 r;
  r.x = o.x + b.x; r.y = o.y + b.y; r.z = o.z + b.z; r.w = o.w + b.w;
  ((float4*)out)[i] = r;
}

// 16 threads per edge; each handles 4 consecutive features.
// Coalesced float4 gather from xw[src], hardware f32 atomic scatter to out[dst].
__global__ __launch_bounds__(256) void scatter_edges(const float* __restrict__ XW,
                                                     const int* __restrict__ src,
                                                     const int* __restrict__ dst,
                                                     const float* __restrict__ ew,
                                                     float* __restrict__ out,
                                                     int nE) {
  const long long t = (long long)blockIdx.x * blockDim.x + threadIdx.x;
  const int e = (int)(t >> 4);
  if (e >= nE) return;
  const int q = (int)(t & 15);

  const float w = ew[e];
  const float4 m = ((const float4*)(XW + (size_t)src[e] * DIM))[q];
  float* o = out + (size_t)dst[e] * DIM + q * 4;
#if defined(__HIP_DEVICE_COMPILE__)
  unsafeAtomicAdd(o + 0, m.x * w);
  unsafeAtomicAdd(o + 1, m.y * w);
  unsafeAtomicAdd(o + 2, m.z * w);
  unsafeAtomicAdd(o + 3, m.w * w);
#endif
}

extern "C" void kernel_launch(void* const* d_in, const int* in_sizes, int n_in,
                              void* d_out, int out_size, void* d_ws, size_t ws_size,
                              hipStream_t stream) {
  const float* in_feat = (const float*)d_in[0];
  const float* ew      = (const float*)d_in[1];
  const float* W1      = (const float*)d_in[2];
  const float* b1      = (const float*)d_in[3];
  const float* W2      = (const float*)d_in[4];
  const float* b2      = (const float*)d_in[5];
  const int*   src     = (const int*)d_in[6];
  const int*   dst     = (const int*)d_in[7];
  const int    nE      = in_sizes[1];

  float* xw   = (float*)d_ws;                       // [NNODES, DIM]
  float* hbuf = xw + (size_t)NNODES * DIM;          // [NNODES, DIM]
  float* outp = (float*)d_out;

  const int gemmBlocks    = (NTILES_M + 7) / 8;                       // 8 waves / block
  const int initBlocks    = (NNODES * (DIM / 4) + 255) / 256;
  const int scatterBlocks = (int)(((long long)nE * 16 + 255) / 256);

  const float* hcur = in_feat;
  for (int step = 0; step < 4; ++step) {
    const float* W = (step == 0) ? W1 : W2;
    const float* b = (step == 0) ? b1 : b2;
    float* tgt = (step == 3) ? outp : hbuf;

    gemm_wmma<<<gemmBlocks, 256, 0, stream>>>(hcur, W, xw);
    init_out<<<initBlocks, 256, 0, stream>>>(in_feat, b, tgt);
    scatter_edges<<<scatterBlocks, 256, 0, stream>>>(xw, src, dst, ew, tgt, nE);

    hcur = tgt;
  }
}